// WhiteTranspose_28406913696445
// MI455X (gfx1250) — compile-verified
//
#include <hip/hip_runtime.h>
#include <hip/hip_bf16.h>

typedef int   v4i __attribute__((ext_vector_type(4)));
typedef float v4f __attribute__((ext_vector_type(4)));

// Problem constants (match reference)
#define BATCH   8
#define IN_CH   64
#define OUT_CH  64
#define LEN     2048
#define NLUT    256

// One block per (o, i) pair. Block = 256 threads (8 wave32s).
// Each block:
//   1) async-copies white_table[o, i, 0:256] (1 KB) into LDS   (CDNA5 async->LDS path)
//   2) streams x[b, i, o, :] -> out[b, o, i, :] for all 8 batches,
//      B128 nontemporal loads/stores, LDS gather for the LUT.
__global__ __launch_bounds__(256)
void WhiteTranspose_28406913696445_kernel(const int* __restrict__ x,
                                          const float* __restrict__ table,
                                          float* __restrict__ out)
{
    __shared__ float lut[NLUT];

    const int tid = threadIdx.x;           // 0..255
    const int o   = blockIdx.x >> 6;       // 0..63
    const int i   = blockIdx.x & 63;       // 0..63

    // ---- CDNA5 async global->LDS copy of the 256-entry LUT slice ----
    {
        const float* gsrc = table + ((size_t)o * IN_CH + i) * NLUT + tid;
        const uint64_t gaddr = (uint64_t)(uintptr_t)gsrc;
        // generic LDS pointer: low 32 bits are the wave-relative LDS byte offset
        const uint32_t laddr = (uint32_t)(uintptr_t)(&lut[tid]);
        asm volatile("global_load_async_to_lds_b32 %0, %1, off"
                     :: "v"(laddr), "v"(gaddr)
                     : "memory");
        asm volatile("s_wait_asynccnt 0x0" ::: "memory");
    }
    __syncthreads();

    // Contiguous 2048-element runs on both sides:
    //   x  [b, i, o, l] : base = b*BS + (i*OUT_CH + o)*LEN
    //   out[b, o, i, l] : base = b*BS + (o*IN_CH  + i)*LEN
    const size_t bstride  = (size_t)IN_CH * OUT_CH * LEN;
    const size_t xin_base = ((size_t)i * OUT_CH + o) * (size_t)LEN;
    const size_t out_base = ((size_t)o * IN_CH  + i) * (size_t)LEN;

#pragma unroll
    for (int b = 0; b < BATCH; ++b) {
        const v4i* __restrict__ src = (const v4i*)(x   + (size_t)b * bstride + xin_base);
        v4f*       __restrict__ dst = (v4f*)      (out + (size_t)b * bstride + out_base);
#pragma unroll
        for (int c = 0; c < LEN / (256 * 4); ++c) {      // 2 chunks of 1024 elems
            const int pos = c * 256 + tid;
            v4i v = __builtin_nontemporal_load(&src[pos]);   // global_load_b128 (NT)
            v4f r;
            r.x = lut[v.x & 0xFF];
            r.y = lut[v.y & 0xFF];
            r.z = lut[v.z & 0xFF];
            r.w = lut[v.w & 0xFF];
            __builtin_nontemporal_store(r, &dst[pos]);       // global_store_b128 (NT)
        }
    }
}

extern "C" void kernel_launch(void* const* d_in, const int* in_sizes, int n_in,
                              void* d_out, int out_size, void* d_ws, size_t ws_size,
                              hipStream_t stream)
{
    (void)in_sizes; (void)n_in; (void)d_ws; (void)ws_size; (void)out_size;
    const int*   x   = (const int*)d_in[0];
    const float* tbl = (const float*)d_in[1];
    float*       out = (float*)d_out;

    dim3 grid(OUT_CH * IN_CH);   // 4096 blocks, one per (o,i)
    dim3 block(256);             // 8 wave32s
    WhiteTranspose_28406913696445_kernel<<<grid, block, 0, stream>>>(x, tbl, out);
}